// _DispatchNeighbors_61787399520646
// MI455X (gfx1250) — compile-verified
//
#include <hip/hip_runtime.h>
#include <stdint.h>

// ---------------------------------------------------------------------------
// DispatchNeighbors for MI455X (gfx1250).
//
// Roofline: ~117 MB streamed index/offset input + ~160 MB streamed output +
// ~3 MB of L2-resident gather tables  =>  ~280 MB HBM => ~12 us @ 23.3 TB/s.
// ~25 FLOPs/pair (0.1 GFLOP total) is ~5 orders of magnitude below any WMMA
// ceiling, and the per-pair 3x3 matvec has a random per-pair molecule id
// (no shared operand), so matrix ops are inapplicable: this is a pure
// bandwidth kernel.
//
// CDNA5-specific paths:
//  * global_load_async_to_lds_b128 + s_wait_asynccnt (ASYNCcnt) to stage the
//    per-molecule cell matrices into LDS once per block.
//  * Nontemporal (TH_NT) hints on all stream-once traffic so L2 keeps only
//    the gather tables.
//  * wave32-sized blocks (256 thr = 8 waves), lane -> pair coalescing.
// All int64 index tables hold values < 2^31, so only the low dword of each
// element is loaded (halves L1 return payload; HBM line traffic unchanged).
// ---------------------------------------------------------------------------

#define BLOCK 256

__global__ __launch_bounds__(BLOCK) void dispatch_neighbors_kernel(
    const float* __restrict__ coords,        // [n_mol*n_atoms, 3] f32
    const float* __restrict__ cell,          // [n_mol, 3, 3] f32
    const int* __restrict__ real_atoms32,    // low dwords of [n_flat] i64
    const int* __restrict__ inv_real32,      // low dwords of [n_flat] i64
    const int* __restrict__ mol_index32,     // low dwords of [n_flat] i64
    const int* __restrict__ pfr32,           // low dwords of [P] i64
    const int* __restrict__ psr32,           // low dwords of [P] i64
    const int* __restrict__ offsets,         // [P, 3] i32
    const int* __restrict__ n_images_p,      // scalar
    float* __restrict__ out,                 // [10*P] f32 (concat outputs)
    int P, int cell_elems)
{
    extern __shared__ float s_cell[];        // n_mol * 9 floats
    const int tid = threadIdx.x;
    const int ni = *n_images_p;              // uniform scalar load (s_load)

    // ---- Stage cell matrices into LDS via CDNA5 async global->LDS copy ----
    {
        const int chunks = cell_elems >> 2;  // 16-byte chunks
        // Low 32 bits of the generic LDS pointer are the LDS byte offset.
        const uint32_t lds_base = (uint32_t)(uintptr_t)&s_cell[0];
        for (int c = tid; c < chunks; c += BLOCK) {
            uint32_t lds_a = lds_base + (uint32_t)c * 16u;
            uint32_t voff  = (uint32_t)c * 16u;
            asm volatile("global_load_async_to_lds_b128 %0, %1, %2"
                         :: "v"(lds_a), "v"(voff), "s"(cell)
                         : "memory");
        }
        // Tail (cell_elems not a multiple of 4) via the normal path.
        for (int e = (chunks << 2) + tid; e < cell_elems; e += BLOCK)
            s_cell[e] = cell[e];
        asm volatile("s_wait_asynccnt 0x0" ::: "memory");
    }
    __syncthreads();

    const int p = blockIdx.x * BLOCK + tid;
    if (p >= P) return;

    const int n_off = 2 * ni + 1;

    // ---- Stream-once inputs: nontemporal, low-dword-only for the i64s ----
    const int pf_raw = __builtin_nontemporal_load(&pfr32[2 * (size_t)p]);
    const int ps_raw = __builtin_nontemporal_load(&psr32[2 * (size_t)p]);
    const int offx = __builtin_nontemporal_load(&offsets[3 * (size_t)p + 0]);
    const int offy = __builtin_nontemporal_load(&offsets[3 * (size_t)p + 1]);
    const int offz = __builtin_nontemporal_load(&offsets[3 * (size_t)p + 2]);

    // ---- Gathers (tables are L2/L1 resident; regular temporal policy) ----
    const int pf = inv_real32[2 * (size_t)pf_raw];
    const int ps = inv_real32[2 * (size_t)ps_raw];

    const int oi = (offz + ni) + n_off * ((offy + ni) + n_off * (offx + ni));

    const int mol = mol_index32[2 * (size_t)pf];
    const float* __restrict__ c = &s_cell[mol * 9];
    const float fx = (float)offx, fy = (float)offy, fz = (float)offz;
    // pair_offsets[j] = sum_i offsets[i] * cell[mol][i][j]
    const float pox = fmaf(fx, c[0], fmaf(fy, c[3], fz * c[6]));
    const float poy = fmaf(fx, c[1], fmaf(fy, c[4], fz * c[7]));
    const float poz = fmaf(fx, c[2], fmaf(fy, c[5], fz * c[8]));

    const int rf = real_atoms32[2 * (size_t)pf];
    const int rs = real_atoms32[2 * (size_t)ps];
    const float* __restrict__ a = &coords[(size_t)rf * 3];
    const float* __restrict__ b = &coords[(size_t)rs * 3];
    const float pcx = a[0] - b[0] + pox;
    const float pcy = a[1] - b[1] + poy;
    const float pcz = a[2] - b[2] + poz;
    const float dist = sqrtf(fmaf(pcx, pcx, fmaf(pcy, pcy, pcz * pcz)));

    // ---- Outputs, concatenated flat in reference return order ----
    const size_t Ps = (size_t)P;
    __builtin_nontemporal_store(dist,      &out[p]);                 // distflat2
    __builtin_nontemporal_store((float)pf, &out[Ps + p]);            // pair_first
    __builtin_nontemporal_store((float)ps, &out[2 * Ps + p]);        // pair_second
    const size_t pc_base = 3 * Ps + 3 * (size_t)p;                   // paircoord
    __builtin_nontemporal_store(pcx, &out[pc_base + 0]);
    __builtin_nontemporal_store(pcy, &out[pc_base + 1]);
    __builtin_nontemporal_store(pcz, &out[pc_base + 2]);
    const size_t of_base = 6 * Ps + 3 * (size_t)p;                   // offsets
    __builtin_nontemporal_store((float)offx, &out[of_base + 0]);
    __builtin_nontemporal_store((float)offy, &out[of_base + 1]);
    __builtin_nontemporal_store((float)offz, &out[of_base + 2]);
    __builtin_nontemporal_store((float)oi,   &out[9 * Ps + p]);      // offset_index
}

extern "C" void kernel_launch(void* const* d_in, const int* in_sizes, int n_in,
                              void* d_out, int out_size, void* d_ws, size_t ws_size,
                              hipStream_t stream) {
    const float* coords     = (const float*)d_in[0];
    const float* cell       = (const float*)d_in[1];
    // int64 tables: values < 2^31, read low dwords only (little-endian).
    const int*   real_atoms = (const int*)d_in[2];
    const int*   inv_real   = (const int*)d_in[3];
    const int*   mol_index  = (const int*)d_in[4];
    const int*   pfr        = (const int*)d_in[5];
    const int*   psr        = (const int*)d_in[6];
    const int*   offsets    = (const int*)d_in[7];
    const int*   n_images   = (const int*)d_in[8];  // first 4B valid for i32/i64-LE
    float*       out        = (float*)d_out;

    const int P          = in_sizes[5];             // number of pairs
    const int cell_elems = in_sizes[1];             // n_mol * 9
    const size_t shmem   = (size_t)cell_elems * sizeof(float);

    if (P <= 0) return;
    dim3 grid((unsigned)((P + BLOCK - 1) / BLOCK));
    dispatch_neighbors_kernel<<<grid, BLOCK, shmem, stream>>>(
        coords, cell, real_atoms, inv_real, mol_index,
        pfr, psr, offsets, n_images, out, P, cell_elems);
}